// ReviewGNN_60438779789899
// MI455X (gfx1250) — compile-verified
//
#include <hip/hip_runtime.h>

// out[i] = dot(diner_table[diner_idx[i]], fcW[0:64])
//        + dot(reviewer_table[reviewer_idx[i]], fcW[64:128]) + fcb
// Pure gather + fp32 dot; GCN layers in the reference are dead code w.r.t. output.
// One wave32 computes 16 outputs via 32x V_WMMA_F32_16X16X4_F32 accumulations.

typedef __attribute__((ext_vector_type(2))) float v2f;
typedef __attribute__((ext_vector_type(8))) float v8f;

__global__ __launch_bounds__(256) void ReviewGNN_head_wmma(
    const int*   __restrict__ diner_idx,
    const int*   __restrict__ reviewer_idx,
    const float* __restrict__ diner_table,     // [NUM_DINERS, 64]
    const float* __restrict__ reviewer_table,  // [NUM_REVIEWERS, 64]
    const float* __restrict__ fcW,             // [128]
    const float* __restrict__ fcb,             // [1]
    float*       __restrict__ out,             // [B]
    int B)
{
    __shared__ float wsh[128];
    const int tid = threadIdx.x;
    if (tid < 128) wsh[tid] = fcW[tid];
    __syncthreads();

    const int lane = tid & 31;
    const int wave = tid >> 5;
    const int m    = lane & 15;   // A-matrix row this lane owns
    const int kh   = lane >> 4;   // K-half: lanes 0-15 -> K%4 in {0,1}, lanes 16-31 -> {2,3}
    const long tile = (long)blockIdx.x * 8 + wave;       // 16 rows per tile
    if (tile * 16 >= B) return;                          // uniform per wave -> EXEC stays all-1

    int row = (int)(tile * 16) + m;
    if (row >= B) row = B - 1;                           // clamp reads for safety (B%16==0 normally)

    const int di = diner_idx[row];
    const int ri = reviewer_idx[row];
    const float* __restrict__ drow = diner_table    + (long)di * 64;
    const float* __restrict__ rrow = reviewer_table + (long)ri * 64;

    // A fragment (16x4 f32): VGPR v of lane L holds x[m, k0 + v + 2*(L>>4)].
    // B fragment (4x16 f32): VGPR v of lane L holds fcW[k0 + v + 2*(L>>4)] for all N columns
    // (replicated across N so every D column carries the dot product).
    const int koff = 2 * kh;
    v8f acc = {};

    #pragma unroll
    for (int k0 = 0; k0 < 64; k0 += 4) {
        v2f a = *(const v2f*)(drow + k0 + koff);
        v2f b = *(const v2f*)(wsh + k0 + koff);
        acc = __builtin_amdgcn_wmma_f32_16x16x4_f32(
            /*neg_a=*/false, a, /*neg_b=*/false, b,
            /*c_mod=*/(short)0, acc, /*reuse_a=*/false, /*reuse_b=*/false);
    }
    #pragma unroll
    for (int k0 = 0; k0 < 64; k0 += 4) {
        v2f a = *(const v2f*)(rrow + k0 + koff);
        v2f b = *(const v2f*)(wsh + 64 + k0 + koff);
        acc = __builtin_amdgcn_wmma_f32_16x16x4_f32(
            false, a, false, b, (short)0, acc, false, false);
    }

    // D layout: acc[r] on lanes 0-15 = D[r, n]; on lanes 16-31 = D[8+r, n-16].
    // Column 0 carriers: lane 0 -> rows 0..7, lane 16 -> rows 8..15.
    const float bias = fcb[0];
    if (m == 0) {
        const long base = tile * 16 + (long)kh * 8;
        if (base + 8 <= (long)B) {
            float4 lo = make_float4(acc[0] + bias, acc[1] + bias, acc[2] + bias, acc[3] + bias);
            float4 hi = make_float4(acc[4] + bias, acc[5] + bias, acc[6] + bias, acc[7] + bias);
            *(float4*)(out + base)     = lo;   // 32B-aligned: base is a multiple of 8 floats
            *(float4*)(out + base + 4) = hi;
        } else {
            #pragma unroll
            for (int r = 0; r < 8; ++r)
                if (base + r < (long)B) out[base + r] = acc[r] + bias;
        }
    }
}

extern "C" void kernel_launch(void* const* d_in, const int* in_sizes, int n_in,
                              void* d_out, int out_size, void* d_ws, size_t ws_size,
                              hipStream_t stream) {
    // setup_inputs() order:
    // 0: diner_idx [B] i32        1: reviewer_idx [B] i32
    // 2: edge_index [2,E] i32     3: diner_table [ND,64] f32
    // 4: reviewer_table [NR,64]   5: W1  6: b1  7: W2  8: b2   (all unused by the output)
    // 9: fc_W [128,1] f32        10: fc_b [1] f32
    const int*   diner_idx      = (const int*)  d_in[0];
    const int*   reviewer_idx   = (const int*)  d_in[1];
    const float* diner_table    = (const float*)d_in[3];
    const float* reviewer_table = (const float*)d_in[4];
    const float* fcW            = (const float*)d_in[9];
    const float* fcb            = (const float*)d_in[10];
    float*       out            = (float*)d_out;

    const int B = in_sizes[0];
    const int tiles  = (B + 15) / 16;       // 16 outputs per wave
    const int blocks = (tiles + 7) / 8;     // 8 waves (256 threads) per block

    ReviewGNN_head_wmma<<<blocks, 256, 0, stream>>>(
        diner_idx, reviewer_idx, diner_table, reviewer_table, fcW, fcb, out, B);
}